// VQC_8632884265186
// MI455X (gfx1250) — compile-verified
//
#include <hip/hip_runtime.h>
#include <math.h>

// ---------------------------------------------------------------------------
// VQC (14 qubits, 4 layers, batch 1024) for MI455X / gfx1250, wave32.
// One 512-thread workgroup (16 waves) per batch element; the full 2^14
// complex state lives in LDS (padded). Rotations fused per-wire, then per-
// 4-wire-group into 16x16 complex unitaries applied with
// V_WMMA_F32_16X16X4_F32. The 14-CNOT ring is fused into a single
// GF(2)-linear amplitude permutation per layer.
// 512 threads (not 1024) so the per-wave VGPR budget is file/4, which avoids
// the scratch spills observed with 8 waves/SIMD.
// ---------------------------------------------------------------------------

typedef __attribute__((ext_vector_type(2))) float v2f;
typedef __attribute__((ext_vector_type(8))) float v8f;

#define NQ        14
#define DIM       16384
#define NLAYERS   4
#define BATCH     1024
#define TPB       512
#define APT       32            // amplitudes per thread (DIM / TPB)
#define PI_F      3.14159265358979323846f

// LDS layout (floats). State arrays are padded: slot(y) = y + (y>>4) so the
// low-bit group pass (row stride 4) doesn't serialize on banks.
#define SZ_STATE  (DIM + (DIM >> 4))          // 17408
#define OFF_SRE   0
#define OFF_SIM   (OFF_SRE + SZ_STATE)        // 17408
#define OFF_U16RE (OFF_SIM + SZ_STATE)        // 34816 : 3 groups * 256
#define OFF_U16IM (OFF_U16RE + 768)
#define OFF_UWRE  (OFF_U16IM + 768)           // 14 wires * 4 entries
#define OFF_UWIM  (OFF_UWRE + 56)
#define OFF_RACC  (OFF_UWIM + 56)             // 15 reduction accumulators
#define SHM_FLOATS (OFF_RACC + 16)            // 36480
#define SHM_BYTES  (SHM_FLOATS * 4)           // 145920 B

__device__ __forceinline__ int slotf(int y) { return y + (y >> 4); }

struct cpx { float r, i; };
__device__ __forceinline__ cpx cmul(cpx a, cpx b) {
  return cpx{a.r * b.r - a.i * b.i, a.r * b.i + a.i * b.r};
}
__device__ __forceinline__ cpx cadd(cpx a, cpx b) { return cpx{a.r + b.r, a.i + b.i}; }

// Apply a 16x16 complex unitary (row-major r*16+c in Ure/Uim) along the 4-bit
// group axis whose row stride in amplitudes is rs (1024 / 64 / 4).
// State viewed as (16 rows) x (1024 cols); 64 column tiles, 4 per wave.
// Complex matmul = 4 real WMMA accumulation chains over 4 K-chunks.
__device__ __forceinline__ void apply_group_wmma(
    float* __restrict__ sRe, float* __restrict__ sIm,
    const float* __restrict__ Ure, const float* __restrict__ Uim,
    int rs, int lane, int waveId)
{
  const int M    = lane & 15;   // A-fragment row / B-N / D-N index
  const int half = lane >> 4;

  // A fragments (uniform across tiles): lane M = lane%16, K = v + 2*half.
  // ain = -Ai for the Re-part accumulation (float WMMA has no A/B NEG).
  v2f ar[4], ai[4], ain[4];
#pragma unroll
  for (int kk = 0; kk < 4; ++kk) {
    const int K0 = kk * 4 + 2 * half;
    v2f r, im;
    r.x  = Ure[M * 16 + K0];     r.y  = Ure[M * 16 + K0 + 1];
    im.x = Uim[M * 16 + K0];     im.y = Uim[M * 16 + K0 + 1];
    ar[kk] = r;  ai[kk] = im;  ain[kk] = -im;
  }

  // Keep the tile loop rolled: full unroll precomputes all B addresses and
  // spills to scratch at this occupancy.
#pragma unroll 1
  for (int t = 0; t < 4; ++t) {
    const int c    = (waveId * 4 + t) * 16 + M;         // this lane's column
    const int base = (c / rs) * (rs * 16) + (c % rs);   // + r*rs gives amp
    v8f accRe = {};
    v8f accIm = {};
#pragma unroll
    for (int kk = 0; kk < 4; ++kk) {
      const int K0 = kk * 4 + 2 * half;
      v2f br, bi;
      br.x = sRe[slotf(base + (K0    ) * rs)];
      br.y = sRe[slotf(base + (K0 + 1) * rs)];
      bi.x = sIm[slotf(base + (K0    ) * rs)];
      bi.y = sIm[slotf(base + (K0 + 1) * rs)];
      // Re += Ar*Br - Ai*Bi ; Im += Ar*Bi + Ai*Br
      accRe = __builtin_amdgcn_wmma_f32_16x16x4_f32(false, ar[kk],  false, br,
                                                    (short)0, accRe, false, false);
      accRe = __builtin_amdgcn_wmma_f32_16x16x4_f32(false, ain[kk], false, bi,
                                                    (short)0, accRe, false, false);
      accIm = __builtin_amdgcn_wmma_f32_16x16x4_f32(false, ar[kk],  false, bi,
                                                    (short)0, accIm, false, false);
      accIm = __builtin_amdgcn_wmma_f32_16x16x4_f32(false, ai[kk],  false, br,
                                                    (short)0, accIm, false, false);
    }
    // D layout: VGPR v -> row (v + 8*half), col = lane%16. All reads of this
    // tile's columns happened above, so in-place store is safe.
#pragma unroll
    for (int v = 0; v < 8; ++v) {
      const int a = slotf(base + (v + 8 * half) * rs);
      sRe[a] = accRe[v];
      sIm[a] = accIm[v];
    }
  }
}

__global__ __launch_bounds__(TPB, 1)
void vqc_kernel(const float* __restrict__ x,     // (1024, 8)
                const float* __restrict__ w_in,  // (14, 8)
                const float* __restrict__ b_in,  // (14)
                const float* __restrict__ w,     // (14, 4, 3)
                const float* __restrict__ w_out, // (1, 14)
                const float* __restrict__ b_out, // (1)
                float* __restrict__ out)         // (1024)
{
  extern __shared__ float smem[];
  float* sRe   = smem + OFF_SRE;
  float* sIm   = smem + OFF_SIM;
  float* U16re = smem + OFF_U16RE;
  float* U16im = smem + OFF_U16IM;
  float* UWre  = smem + OFF_UWRE;
  float* UWim  = smem + OFF_UWIM;
  float* rAcc  = smem + OFF_RACC;

  const int tid    = threadIdx.x;
  const int lane   = tid & 31;
  const int waveId = tid >> 5;
  const int b      = blockIdx.x;

  if (tid < 15) rAcc[tid] = 0.0f;

  // ---- Input layer (uniform per workgroup) + RY encoding folded into an
  //      analytic product-state init: amp(y) = prod_bit (bit ? sin : cos).
  //      Wire i is the MSB-side qubit: bit index = 13 - i.
  float cb[NQ], sb[NQ];
#pragma unroll
  for (int i = 0; i < NQ; ++i) {
    float a = b_in[i];
#pragma unroll
    for (int k = 0; k < 8; ++k) a += x[b * 8 + k] * w_in[i * 8 + k];
    const float h = 0.5f * PI_F * a;        // (pi * input)/2
    cb[13 - i] = cosf(h);
    sb[13 - i] = sinf(h);
  }
#pragma unroll 1
  for (int k = 0; k < APT; ++k) {
    const int y = tid * APT + k;
    float p = 1.0f;
#pragma unroll
    for (int bit = 0; bit < NQ; ++bit) p *= ((y >> bit) & 1) ? sb[bit] : cb[bit];
    const int a = slotf(y);
    sRe[a] = p;
    sIm[a] = 0.0f;
  }
  __syncthreads();

  for (int j = 0; j < NLAYERS; ++j) {
    // ---- Fused per-wire unitary U = RZ(t3) * RY(t2) * RX(t1) (uniform)
    if (tid < NQ) {
      const int i = tid;
      const float t1 = w[(i * 4 + j) * 3 + 0];
      const float t2 = w[(i * 4 + j) * 3 + 1];
      const float t3 = w[(i * 4 + j) * 3 + 2];
      const float c1 = cosf(0.5f * t1), s1 = sinf(0.5f * t1);
      const float c2 = cosf(0.5f * t2), s2 = sinf(0.5f * t2);
      const float cz = cosf(0.5f * t3), sz = sinf(0.5f * t3);
      cpx RX[2][2] = {{{c1, 0.f}, {0.f, -s1}}, {{0.f, -s1}, {c1, 0.f}}};
      cpx RY[2][2] = {{{c2, 0.f}, {-s2, 0.f}}, {{s2, 0.f}, {c2, 0.f}}};
      const cpx ez[2] = {{cz, -sz}, {cz, sz}};   // RZ diagonal
#pragma unroll
      for (int r = 0; r < 2; ++r)
#pragma unroll
        for (int c = 0; c < 2; ++c) {
          cpx m = cadd(cmul(RY[r][0], RX[0][c]), cmul(RY[r][1], RX[1][c]));
          cpx u = cmul(ez[r], m);
          UWre[i * 4 + r * 2 + c] = u.r;
          UWim[i * 4 + r * 2 + c] = u.i;
        }
    }
    __syncthreads();

    // ---- Build the three 16x16 Kronecker unitaries (wires 4g..4g+3).
    //      Row bit (3-q) of the group axis == wire 4g+q.
    for (int e2 = tid; e2 < 768; e2 += TPB) {
      const int g = e2 >> 8, e = e2 & 255, r = e >> 4, c = e & 15;
      cpx p = {1.f, 0.f};
#pragma unroll
      for (int q = 0; q < 4; ++q) {
        const int wi = g * 4 + q;
        const int rb = (r >> (3 - q)) & 1, cbit = (c >> (3 - q)) & 1;
        cpx u = {UWre[wi * 4 + rb * 2 + cbit], UWim[wi * 4 + rb * 2 + cbit]};
        p = cmul(p, u);
      }
      U16re[g * 256 + e] = p.r;
      U16im[g * 256 + e] = p.i;
    }
    __syncthreads();

    // ---- WMMA group passes: bits 13..10 (rs=1024), 9..6 (rs=64), 5..2 (rs=4)
    apply_group_wmma(sRe, sIm, U16re + 0,   U16im + 0,   1024, lane, waveId);
    __syncthreads();
    apply_group_wmma(sRe, sIm, U16re + 256, U16im + 256, 64,   lane, waveId);
    __syncthreads();
    apply_group_wmma(sRe, sIm, U16re + 512, U16im + 512, 4,    lane, waveId);
    __syncthreads();

    // ---- Scalar 4x4 pass: wires 12,13 (bits 1,0). U4 = U12 (x) U13.
    {
      cpx u4[4][4];
#pragma unroll
      for (int e = 0; e < 4; ++e)
#pragma unroll
        for (int f = 0; f < 4; ++f) {
          cpx a12 = {UWre[12 * 4 + ((e >> 1) & 1) * 2 + ((f >> 1) & 1)],
                     UWim[12 * 4 + ((e >> 1) & 1) * 2 + ((f >> 1) & 1)]};
          cpx a13 = {UWre[13 * 4 + (e & 1) * 2 + (f & 1)],
                     UWim[13 * 4 + (e & 1) * 2 + (f & 1)]};
          u4[e][f] = cmul(a12, a13);
        }
#pragma unroll 1
      for (int m = 0; m < 8; ++m) {
        const int base = (tid * 8 + m) * 4;     // 4 consecutive amplitudes
        float xr[4], xi[4];
#pragma unroll
        for (int e = 0; e < 4; ++e) {
          const int a = slotf(base + e);
          xr[e] = sRe[a]; xi[e] = sIm[a];
        }
#pragma unroll
        for (int e = 0; e < 4; ++e) {
          float nr = 0.f, ni = 0.f;
#pragma unroll
          for (int f = 0; f < 4; ++f) {
            nr += u4[e][f].r * xr[f] - u4[e][f].i * xi[f];
            ni += u4[e][f].r * xi[f] + u4[e][f].i * xr[f];
          }
          const int a = slotf(base + e);
          sRe[a] = nr; sIm[a] = ni;
        }
      }
    }
    __syncthreads();

    // ---- Fused CNOT ring: C(0,1)..C(12,13),C(13,0) is one GF(2)-linear
    //      permutation: new[y] = old[g(y)], g = f_first(...f_last(y)).
    //      Wire w -> bit 13-w. Gather to regs, barrier, scatter.
    {
      float pr[APT], pi[APT];
#pragma unroll
      for (int k = 0; k < APT; ++k) {
        const int y = tid * APT + k;
        int z = y;
        z ^= ((z >> 0) & 1) << 13;                    // C(13,0): bc=0, bt=13
#pragma unroll
        for (int kg = 13; kg >= 1; --kg)              // C(kg-1, kg)
          z ^= ((z >> (14 - kg)) & 1) << (13 - kg);
        const int a = slotf(z);
        pr[k] = sRe[a]; pi[k] = sIm[a];
      }
      __syncthreads();                                // all reads done
#pragma unroll
      for (int k = 0; k < APT; ++k) {
        const int a = slotf(tid * APT + k);
        sRe[a] = pr[k]; sIm[a] = pi[k];
      }
      __syncthreads();
    }
  }

  // ---- <Z_i> reduction + output layer.  z_i = tot - 2 * sum_{bit(13-i)=1} p.
  float vals[15];
#pragma unroll
  for (int v = 0; v < 15; ++v) vals[v] = 0.f;
#pragma unroll 1
  for (int k = 0; k < APT; ++k) {
    const int y = tid * APT + k;
    const int a = slotf(y);
    const float re = sRe[a], im = sIm[a];
    const float p = re * re + im * im;
    vals[14] += p;
#pragma unroll
    for (int bit = 0; bit < NQ; ++bit) vals[bit] += ((y >> bit) & 1) ? p : 0.f;
  }
#pragma unroll
  for (int s = 16; s > 0; s >>= 1) {
#pragma unroll
    for (int v = 0; v < 15; ++v) vals[v] += __shfl_xor(vals[v], s, 32);
  }
  if (lane == 0) {
#pragma unroll
    for (int v = 0; v < 15; ++v) atomicAdd(&rAcc[v], vals[v]);
  }
  __syncthreads();

  if (tid == 0) {
    const float tot = rAcc[14];
    float o = b_out[0];
#pragma unroll
    for (int i = 0; i < NQ; ++i) o += w_out[i] * (tot - 2.0f * rAcc[13 - i]);
    out[b] = o;
  }
}

extern "C" void kernel_launch(void* const* d_in, const int* in_sizes, int n_in,
                              void* d_out, int out_size, void* d_ws, size_t ws_size,
                              hipStream_t stream) {
  const float* x     = (const float*)d_in[0];
  const float* w_in  = (const float*)d_in[1];
  const float* b_in  = (const float*)d_in[2];
  const float* wts   = (const float*)d_in[3];
  const float* w_out = (const float*)d_in[4];
  const float* b_out = (const float*)d_in[5];
  float* out = (float*)d_out;

  (void)in_sizes; (void)n_in; (void)out_size; (void)d_ws; (void)ws_size;

  // 142.5 KB dynamic LDS per workgroup.
  hipFuncSetAttribute((const void*)vqc_kernel,
                      hipFuncAttributeMaxDynamicSharedMemorySize, SHM_BYTES);

  vqc_kernel<<<BATCH, TPB, SHM_BYTES, stream>>>(x, w_in, b_in, wts, w_out, b_out, out);
}